// DistributedMPNN_40638980554894
// MI455X (gfx1250) — compile-verified
//
#include <hip/hip_runtime.h>
#include <math.h>

typedef __attribute__((ext_vector_type(2))) float v2f;
typedef __attribute__((ext_vector_type(8))) float v8f;

#define NK   8192
#define DEG  32
#define GE   8
#define WAVES 8
#define LDSF 1152   // floats per wave region: hbuf 32x33=1056 | aggr 32 | u 16 | comb 8 | pad

__device__ __forceinline__ v8f wmma_f32(v2f a, v2f b, v8f c) {
  // V_WMMA_F32_16X16X4_F32: D(16x16) = A(16x4) * B(4x16) + C
  return __builtin_amdgcn_wmma_f32_16x16x4_f32(false, a, false, b, (short)0, c, false, false);
}

__global__ void __launch_bounds__(256, 1) mpnn_p_kernel(
    const float* __restrict__ x,   const float* __restrict__ edge_attr,
    const float* __restrict__ Wm1, const float* __restrict__ bm1,
    const float* __restrict__ Wm2, const float* __restrict__ bm2,
    const float* __restrict__ Wu1, const float* __restrict__ bu1,
    const float* __restrict__ Wu2, const float* __restrict__ bu2,
    const float* __restrict__ Wh1, const float* __restrict__ bh1,
    const float* __restrict__ Wh2, const float* __restrict__ bh2,
    float* __restrict__ p_out) {
  __shared__ float smem[WAVES * LDSF];

  const int lane = threadIdx.x & 31;
  const int wid  = threadIdx.x >> 5;
  const int k    = blockIdx.x * WAVES + wid;

  float* hb   = &smem[wid * LDSF];   // 32 rows x stride 33
  float* agg  = hb + 1056;           // 32 floats
  float* ubuf = hb + 1088;           // 16 floats
  float* cbuf = hb + 1104;           // 8 floats

  const int l15   = lane & 15;
  const int hi    = lane >> 4;       // 0 | 1
  const int koff2 = hi << 1;         // 0 | 2

  const float* Wm1k = Wm1 + (size_t)k * (2 + GE) * 32;
  const float* bm1k = bm1 + (size_t)k * 32;
  const float* Wm2k = Wm2 + (size_t)k * 32 * 32;
  const float* bm2k = bm2 + (size_t)k * 32;
  const float* Wu1k = Wu1 + (size_t)k * (33 + GE) * 16;
  const float* bu1k = bu1 + (size_t)k * 16;
  const float* Wu2k = Wu2 + (size_t)k * 16 * GE;
  const float* bu2k = bu2 + (size_t)k * GE;
  const float* Wh1k = Wh1 + (size_t)k * GE * 16;
  const float* bh1k = bh1 + (size_t)k * 16;
  const float* Wh2k = Wh2 + (size_t)k * 16;
  const float* bh2k = bh2 + (size_t)k;

  // ---- node state (replicated across lanes) and per-lane edge attrs ----
  float xr[1 + GE];
#pragma unroll
  for (int d = 0; d < 1 + GE; ++d) xr[d] = x[(size_t)k * (1 + GE) + d];
  const float ea0 = edge_attr[(size_t)k * DEG + l15];        // rows 0..15
  const float ea1 = edge_attr[(size_t)k * DEG + 16 + l15];   // rows 16..31

  // ---- preload B fragments (layer-invariant, kept in VGPRs) ----
  // B 4x16 f32 layout (mirror of A): lane<16: v0=B[k0+0][n], v1=B[k0+1][n];
  //                                  lane>=16: v0=B[k0+2][n], v1=B[k0+3][n]
  v2f bWm1f[3][2];
#pragma unroll
  for (int ks = 0; ks < 3; ++ks) {
    const int kb = ks * 4 + koff2;
#pragma unroll
    for (int nt = 0; nt < 2; ++nt) {
      const int n = nt * 16 + l15;
      bWm1f[ks][nt].x = (kb     < 10) ? Wm1k[kb * 32 + n]       : 0.f;
      bWm1f[ks][nt].y = (kb + 1 < 10) ? Wm1k[(kb + 1) * 32 + n] : 0.f;
    }
  }
  v2f bWm2f[8][2];
#pragma unroll
  for (int ks = 0; ks < 8; ++ks) {
    const int kb = ks * 4 + koff2;
#pragma unroll
    for (int nt = 0; nt < 2; ++nt) {
      const int n = nt * 16 + l15;
      bWm2f[ks][nt].x = Wm2k[kb * 32 + n];
      bWm2f[ks][nt].y = Wm2k[(kb + 1) * 32 + n];
    }
  }

  // ================= 3 conv layers =================
  for (int layer = 0; layer < 3; ++layer) {
    // ---- GEMM1: tmp(32x10) @ Wm1(10x32) -> h in LDS ----
#pragma unroll
    for (int mt = 0; mt < 2; ++mt) {
      const float eaRow = mt ? ea1 : ea0;
      // A fragments built in registers; tmp[row][0..8]=x[k], tmp[row][9]=edge_attr
      v2f aT[3];
      aT[0].x = hi ? xr[2] : xr[0];  aT[0].y = hi ? xr[3] : xr[1];
      aT[1].x = hi ? xr[6] : xr[4];  aT[1].y = hi ? xr[7] : xr[5];
      aT[2].x = hi ? 0.f   : xr[8];  aT[2].y = hi ? 0.f   : eaRow;
#pragma unroll
      for (int nt = 0; nt < 2; ++nt) {
        v8f c = {};
        c = wmma_f32(aT[0], bWm1f[0][nt], c);
        c = wmma_f32(aT[1], bWm1f[1][nt], c);
        c = wmma_f32(aT[2], bWm1f[2][nt], c);
        const float bias  = bm1k[nt * 16 + l15];
        const int   mbase = mt * 16 + hi * 8;
#pragma unroll
        for (int r = 0; r < 8; ++r)
          hb[(mbase + r) * 33 + nt * 16 + l15] = fmaxf(c[r] + bias, 0.f);
      }
    }
    __syncthreads();

    // ---- GEMM2: h(32x32) @ Wm2(32x32), ReLU, row-sum -> aggr(32) ----
    float cs0 = 0.f, cs1 = 0.f;
#pragma unroll
    for (int mt = 0; mt < 2; ++mt) {
      const int row = mt * 16 + l15;
      v2f aH[8];
#pragma unroll
      for (int ks = 0; ks < 8; ++ks) {
        const int kk = ks * 4 + koff2;
        aH[ks].x = hb[row * 33 + kk];
        aH[ks].y = hb[row * 33 + kk + 1];
      }
#pragma unroll
      for (int nt = 0; nt < 2; ++nt) {
        v8f c = {};
#pragma unroll
        for (int ks = 0; ks < 8; ++ks) c = wmma_f32(aH[ks], bWm2f[ks][nt], c);
        const float bias = bm2k[nt * 16 + l15];
        float s = 0.f;
#pragma unroll
        for (int r = 0; r < 8; ++r) s += fmaxf(c[r] + bias, 0.f);
        if (nt == 0) cs0 += s; else cs1 += s;
      }
    }
    cs0 += __shfl_xor(cs0, 16, 32);
    cs1 += __shfl_xor(cs1, 16, 32);
    if (lane < 16) { agg[l15] = cs0; agg[16 + l15] = cs1; }
    __syncthreads();

    // ---- u = relu(t(41) @ Wu1(41x16) + bu1), lanes 0..15 ----
    if (lane < 16) {
      float acc = bu1k[lane];
#pragma unroll
      for (int d = 0; d < 1 + GE; ++d) acc += xr[d] * Wu1k[d * 16 + lane];
      for (int d = 1 + GE; d < 33 + GE; ++d) acc += agg[d - (1 + GE)] * Wu1k[d * 16 + lane];
      ubuf[lane] = fmaxf(acc, 0.f);
    }
    __syncthreads();

    // ---- comb = relu(u(16) @ Wu2(16x8) + bu2), lanes 0..7 ----
    if (lane < GE) {
      float acc = bu2k[lane];
#pragma unroll
      for (int h = 0; h < 16; ++h) acc += ubuf[h] * Wu2k[h * GE + lane];
      cbuf[lane] = fmaxf(acc, 0.f);
    }
    __syncthreads();

    // ---- new x = [x[:,0], comb] broadcast to all lanes ----
#pragma unroll
    for (int d = 1; d < 1 + GE; ++d) xr[d] = cbuf[d - 1];
    __syncthreads();
  }

  // ================= head: emb(8) -> 16 -> sigmoid =================
  float val = 0.f;
  if (lane < 16) {
    float acc = bh1k[lane];
#pragma unroll
    for (int d = 0; d < GE; ++d) acc += xr[1 + d] * Wh1k[d * 16 + lane];
    val = fmaxf(acc, 0.f) * Wh2k[lane];
  }
  val += __shfl_xor(val, 16, 32);
  val += __shfl_xor(val, 8, 32);
  val += __shfl_xor(val, 4, 32);
  val += __shfl_xor(val, 2, 32);
  val += __shfl_xor(val, 1, 32);
  if (lane == 0) {
    const float s = val + bh2k[0];
    p_out[k] = 1.f / (1.f + expf(-s));
  }
}

__global__ void __launch_bounds__(256, 1) mpnn_rate_kernel(
    const float* __restrict__ H, const int* __restrict__ dst,
    const float* __restrict__ p, float* __restrict__ out) {
  const int lane = threadIdx.x & 31;
  const int wid  = threadIdx.x >> 5;
  const int k    = blockIdx.x * WAVES + wid;

  const int d = dst[k * DEG + lane];
  float contrib = p[d] * H[(size_t)k * NK + d];
#pragma unroll
  for (int off = 16; off >= 1; off >>= 1) contrib += __shfl_xor(contrib, off, 32);
  if (lane == 0) {
    const float VARC = 6.2946270589708374e-15f;  // 10^-19.9 * 5e6 / 10
    const float interf = contrib + VARC;
    const float valid  = p[k] * H[(size_t)k * NK + k];
    out[k] = -log2f(1.f + valid / interf);
  }
}

extern "C" void kernel_launch(void* const* d_in, const int* in_sizes, int n_in,
                              void* d_out, int out_size, void* d_ws, size_t ws_size,
                              hipStream_t stream) {
  (void)in_sizes; (void)n_in; (void)out_size; (void)ws_size;
  const float* x         = (const float*)d_in[0];
  const float* edge_attr = (const float*)d_in[1];
  const float* H         = (const float*)d_in[2];
  const int*   edge_idx  = (const int*)  d_in[3];
  const float* Wm1 = (const float*)d_in[4];
  const float* bm1 = (const float*)d_in[5];
  const float* Wm2 = (const float*)d_in[6];
  const float* bm2 = (const float*)d_in[7];
  const float* Wu1 = (const float*)d_in[8];
  const float* bu1 = (const float*)d_in[9];
  const float* Wu2 = (const float*)d_in[10];
  const float* bu2 = (const float*)d_in[11];
  const float* Wh1 = (const float*)d_in[12];
  const float* bh1 = (const float*)d_in[13];
  const float* Wh2 = (const float*)d_in[14];
  const float* bh2 = (const float*)d_in[15];

  float*      p_ws = (float*)d_ws;             // K floats of scratch
  const int*  dst  = edge_idx + NK * DEG;      // edge_index row 1

  dim3 block(256);
  dim3 grid(NK / WAVES);
  mpnn_p_kernel<<<grid, block, 0, stream>>>(x, edge_attr, Wm1, bm1, Wm2, bm2,
                                            Wu1, bu1, Wu2, bu2, Wh1, bh1, Wh2, bh2, p_ws);
  mpnn_rate_kernel<<<grid, block, 0, stream>>>(H, dst, p_ws, (float*)d_out);
}